// AttnDecoder_11510512353401
// MI455X (gfx1250) — compile-verified
//
#include <hip/hip_runtime.h>
#include <hip/hip_bf16.h>

typedef __attribute__((ext_vector_type(2))) float v2f;
typedef __attribute__((ext_vector_type(8))) float v8f;

// ---------------------------------------------------------------------------
// K1: emb gather + hdot[b] = hidden[b] . energy_W[0:H] + energy_b
//     also copies emb and hidden into the fused RNN-input A matrix [64 x 3584]
//     A layout per row b: [context(2048) | emb(512) | hidden(1024)]
// ---------------------------------------------------------------------------
__global__ __launch_bounds__(256) void embed_hdot_kernel(
    const int* __restrict__ x, const float* __restrict__ hidden,
    const float* __restrict__ emb_table, const float* __restrict__ eW,
    const float* __restrict__ eb, float* __restrict__ hdot,
    float* __restrict__ Arnn) {
  const int b = blockIdx.x;
  const int t = threadIdx.x;
  __shared__ float red[256];
  const float* hb = hidden + b * 1024;
  float s = 0.f;
#pragma unroll
  for (int i = 0; i < 4; ++i) s += hb[i * 256 + t] * eW[i * 256 + t];
  red[t] = s;
  __syncthreads();
  for (int k = 128; k > 0; k >>= 1) {
    if (t < k) red[t] += red[t + k];
    __syncthreads();
  }
  if (t == 0) hdot[b] = red[0] + eb[0];
  const int row = x[b];
#pragma unroll
  for (int i = 0; i < 2; ++i)
    Arnn[b * 3584 + 2048 + i * 256 + t] =
        emb_table[(size_t)row * 512 + i * 256 + t];
#pragma unroll
  for (int i = 0; i < 4; ++i)
    Arnn[b * 3584 + 2560 + i * 256 + t] = hb[i * 256 + t];
}

// ---------------------------------------------------------------------------
// K2: energy[s,b] = relu(hdot[b] + enc[s,b,:] . energy_W[H:3H])
//     one wave32 per (s,b), float4 streaming of the 8KB row
// ---------------------------------------------------------------------------
__global__ __launch_bounds__(256) void energy_kernel(
    const float* __restrict__ enc, const float* __restrict__ eW,
    const float* __restrict__ hdot, float* __restrict__ energy) {
  __shared__ float4 wl4[512];  // energy_W[H..3H]
  float* wl = (float*)wl4;
  for (int i = threadIdx.x; i < 2048; i += 256) wl[i] = eW[1024 + i];
  __syncthreads();
  const int wave = threadIdx.x >> 5, lane = threadIdx.x & 31;
  const int g = blockIdx.x * 8 + wave;  // 0..32767
  const int s = g >> 6, b = g & 63;
  const float4* row = (const float4*)(enc + ((size_t)s * 64 + b) * 2048);
  float sum = 0.f;
#pragma unroll
  for (int i = 0; i < 16; ++i) {
    float4 e = row[i * 32 + lane];
    float4 w = wl4[i * 32 + lane];
    sum += e.x * w.x + e.y * w.y + e.z * w.z + e.w * w.w;
  }
#pragma unroll
  for (int off = 16; off > 0; off >>= 1) sum += __shfl_xor(sum, off, 32);
  if (lane == 0) {
    float v = hdot[b] + sum;
    energy[s * 64 + b] = v > 0.f ? v : 0.f;
  }
}

// ---------------------------------------------------------------------------
// K3: in-place softmax over seq dim (512) per batch column b
// ---------------------------------------------------------------------------
__global__ __launch_bounds__(256) void softmax_kernel(float* __restrict__ e) {
  const int b = blockIdx.x;
  const int t = threadIdx.x;
  __shared__ float red[256];
  float v0 = e[t * 64 + b];
  float v1 = e[(t + 256) * 64 + b];
  red[t] = fmaxf(v0, v1);
  __syncthreads();
  for (int k = 128; k > 0; k >>= 1) {
    if (t < k) red[t] = fmaxf(red[t], red[t + k]);
    __syncthreads();
  }
  const float mx = red[0];
  __syncthreads();
  float e0 = expf(v0 - mx), e1 = expf(v1 - mx);
  red[t] = e0 + e1;
  __syncthreads();
  for (int k = 128; k > 0; k >>= 1) {
    if (t < k) red[t] += red[t + k];
    __syncthreads();
  }
  const float inv = 1.f / red[0];
  e[t * 64 + b] = e0 * inv;
  e[(t + 256) * 64 + b] = e1 * inv;
}

// ---------------------------------------------------------------------------
// K4: context[b,d] = sum_s attn[s,b] * enc[s,b,d]  (written into Arnn[b,0:2048])
//     block = (b, 256-wide d tile); deterministic serial reduction over s
// ---------------------------------------------------------------------------
__global__ __launch_bounds__(256) void context_kernel(
    const float* __restrict__ enc, const float* __restrict__ attn,
    float* __restrict__ Arnn) {
  const int b = blockIdx.x >> 3;
  const int dt = blockIdx.x & 7;
  __shared__ float a[512];
  a[threadIdx.x] = attn[threadIdx.x * 64 + b];
  a[threadIdx.x + 256] = attn[(threadIdx.x + 256) * 64 + b];
  __syncthreads();
  const float* base = enc + (size_t)b * 2048 + dt * 256 + threadIdx.x;
  float acc = 0.f;
#pragma unroll 4
  for (int s = 0; s < 512; ++s) {
    if (s + 8 < 512)
      __builtin_prefetch(base + (size_t)(s + 8) * 131072, 0, 0);
    acc += a[s] * base[(size_t)s * 131072];  // stride = B*2H
  }
  Arnn[b * 3584 + dt * 256 + threadIdx.x] = acc;
}

// ---------------------------------------------------------------------------
// K5/K7: f32 WMMA GEMM   C[64,N] = A[64,K] * W[N,K]^T + bias1 (+bias2)
//   W element (n,k) = k<k1 ? W1[n*ldw1+k] : W2[n*ldw2+k-k1]  (fused W_ih|W_hh)
//   block: 256 thr = 8 waves = 4 m-tiles x 2 n-tiles of 16x16; K chunked by 32.
//   Software-pipelined: global fetch of chunk c+1 overlaps WMMA on chunk c,
//   with double-buffered LDS. Tile rows padded to 34 floats so each fragment
//   (two adjacent k values) is one 8B-aligned ds_load_b64, conflict-free
//   (stride 34 across 16 lanes -> 17*i mod 32 is a permutation).
//   V_WMMA_F32_16X16X4_F32 layouts: A: lane m=L%16, k=2*(L/16)+v;
//   B: lane n=L%16, k=2*(L/16)+v; D: lane n=L%16, m=v+8*(L/16).
// ---------------------------------------------------------------------------
__global__ __launch_bounds__(256) void gemm_wmma_kernel(
    const float* __restrict__ A, int lda, const float* __restrict__ W1,
    int ldw1, int k1, const float* __restrict__ W2, int ldw2,
    const float* __restrict__ bias1, const float* __restrict__ bias2,
    float* __restrict__ C, int ldc, int K) {
  __shared__ float As[2][64][34];
  __shared__ float Ws[2][32][34];
  const int t = threadIdx.x;
  const int n0 = blockIdx.x * 32;
  const int wave = t >> 5, lane = t & 31;
  const int mt = wave & 3, nt = wave >> 2;
  const int lm = lane & 15, half = lane >> 4;

  float aReg[8], wReg[4];
  auto loadChunk = [&](int k0) {
#pragma unroll
    for (int r = 0; r < 8; ++r) {  // A chunk 64x32, coalesced (one row / wave)
      int idx = r * 256 + t;
      int row = idx >> 5, kk = idx & 31;
      aReg[r] = A[(size_t)row * lda + k0 + kk];
    }
    const bool first = (k0 < k1);
#pragma unroll
    for (int r = 0; r < 4; ++r) {  // W chunk 32x32, coalesced
      int idx = r * 256 + t;
      int n = idx >> 5, kk = idx & 31;
      int gk = k0 + kk;
      wReg[r] = first ? W1[(size_t)(n0 + n) * ldw1 + gk]
                      : W2[(size_t)(n0 + n) * ldw2 + (gk - k1)];
    }
  };
  auto storeChunk = [&](int buf) {
#pragma unroll
    for (int r = 0; r < 8; ++r) {
      int idx = r * 256 + t;
      As[buf][idx >> 5][idx & 31] = aReg[r];
    }
#pragma unroll
    for (int r = 0; r < 4; ++r) {
      int idx = r * 256 + t;
      Ws[buf][idx >> 5][idx & 31] = wReg[r];
    }
  };

  v8f acc = {};
  const int nc = K >> 5;
  loadChunk(0);
  storeChunk(0);
  for (int c = 0; c < nc; ++c) {
    __syncthreads();  // LDS buf[c&1] ready; buf[1-(c&1)] fully consumed
    if (c + 1 < nc) loadChunk((c + 1) << 5);  // global loads in flight...
    const int buf = c & 1;
    v2f af[8], bf[8];
#pragma unroll
    for (int kk = 0; kk < 8; ++kk) {  // ...while fragments stream from LDS
      const int kb = kk * 4 + half * 2;
      af[kk] = *(const v2f*)&As[buf][mt * 16 + lm][kb];
      bf[kk] = *(const v2f*)&Ws[buf][nt * 16 + lm][kb];
    }
#pragma unroll
    for (int kk = 0; kk < 8; ++kk)  // ...and WMMAs issue back-to-back
      acc = __builtin_amdgcn_wmma_f32_16x16x4_f32(false, af[kk], false, bf[kk],
                                                  (short)0, acc, false, false);
    if (c + 1 < nc) storeChunk((c + 1) & 1);
  }
  const int col = n0 + nt * 16 + lm;
  const float bsum = bias1[col] + (bias2 ? bias2[col] : 0.f);
#pragma unroll
  for (int v = 0; v < 8; ++v) {
    int row = mt * 16 + v + 8 * half;
    C[(size_t)row * ldc + col] = acc[v] + bsum;
  }
}

// ---------------------------------------------------------------------------
// K6: LSTM pointwise: gates[64,4096] -> h_new, c_new
// ---------------------------------------------------------------------------
__global__ __launch_bounds__(256) void lstm_kernel(
    const float* __restrict__ gates, const float* __restrict__ cell,
    float* __restrict__ hnew_ws, float* __restrict__ out_h,
    float* __restrict__ out_c) {
  const int idx = blockIdx.x * 256 + threadIdx.x;  // 65536
  const int b = idx >> 10, h = idx & 1023;
  const float* g = gates + b * 4096;
  float i_ = g[h], f_ = g[1024 + h], gg = g[2048 + h], o_ = g[3072 + h];
  i_ = 1.f / (1.f + expf(-i_));
  f_ = 1.f / (1.f + expf(-f_));
  o_ = 1.f / (1.f + expf(-o_));
  gg = tanhf(gg);
  float c = f_ * cell[idx] + i_ * gg;
  float hn = o_ * tanhf(c);
  hnew_ws[idx] = hn;
  out_h[idx] = hn;
  out_c[idx] = c;
}

extern "C" void kernel_launch(void* const* d_in, const int* in_sizes, int n_in,
                              void* d_out, int out_size, void* d_ws,
                              size_t ws_size, hipStream_t stream) {
  const int* x = (const int*)d_in[0];
  const float* enc = (const float*)d_in[1];
  const float* hidden = (const float*)d_in[2];
  const float* cell = (const float*)d_in[3];
  const float* emb_table = (const float*)d_in[4];
  const float* energy_W = (const float*)d_in[5];
  const float* energy_b = (const float*)d_in[6];
  const float* W_ih = (const float*)d_in[7];
  const float* W_hh = (const float*)d_in[8];
  const float* b_ih = (const float*)d_in[9];
  const float* b_hh = (const float*)d_in[10];
  const float* fc_W = (const float*)d_in[11];
  const float* fc_b = (const float*)d_in[12];

  float* out = (float*)d_out;
  float* preds = out;                    // 64*32000
  float* out_h = out + 2048000;          // 64*1024
  float* out_c = out + 2048000 + 65536;  // 64*1024

  float* ws = (float*)d_ws;
  float* energy = ws;           // 32768
  float* hdot = ws + 32768;     // 64
  float* Arnn = ws + 32832;     // 64*3584 = 229376
  float* gatesb = ws + 262208;  // 64*4096 = 262144
  float* hnew = ws + 524352;    // 65536  (total ~2.4 MB)

  embed_hdot_kernel<<<64, 256, 0, stream>>>(x, hidden, emb_table, energy_W,
                                            energy_b, hdot, Arnn);
  energy_kernel<<<4096, 256, 0, stream>>>(enc, energy_W, hdot, energy);
  softmax_kernel<<<64, 256, 0, stream>>>(energy);
  context_kernel<<<512, 256, 0, stream>>>(enc, energy, Arnn);
  gemm_wmma_kernel<<<128, 256, 0, stream>>>(Arnn, 3584, W_ih, 2560, 2560, W_hh,
                                            1024, b_ih, b_hh, gatesb, 4096,
                                            3584);
  lstm_kernel<<<256, 256, 0, stream>>>(gatesb, cell, hnew, out_h, out_c);
  gemm_wmma_kernel<<<1000, 256, 0, stream>>>(hnew, 1024, fc_W, 1024, 0x40000000,
                                             fc_W, 1024, fc_b, nullptr, preds,
                                             32000, 1024);
}